// GramSmidth_35811437314698
// MI455X (gfx1250) — compile-verified
//
#include <hip/hip_runtime.h>

#define THREADS       256
#define TILE_MATS     1024                      // matrices per block tile
#define TILE_FLOATS   (TILE_MATS * 9)           // 9216 floats = 36 KB LDS
#define F4_PER_TILE   (TILE_FLOATS / 4)         // 2304 float4
#define F4_PER_THREAD (F4_PER_TILE / THREADS)   // 9
#define MATS_PER_THR  (TILE_MATS / THREADS)     // 4
#define CHUNK_BYTES   (THREADS * 16)            // 4096 B per unrolled step (fits 24-bit IOFFSET)

typedef int v4i __attribute__((ext_vector_type(4)));
typedef __attribute__((address_space(1))) v4i glob_v4i;  // global (AS1)
typedef __attribute__((address_space(3))) v4i lds_v4i;   // LDS (AS3)

#if defined(__AMDGCN__) && \
    __has_builtin(__builtin_amdgcn_global_load_async_to_lds_b128) && \
    __has_builtin(__builtin_amdgcn_global_store_async_from_lds_b128)
#define GS_HAS_ASYNC_LDS 1
#else
#define GS_HAS_ASYNC_LDS 0
#endif

__device__ __forceinline__ void gs_wait_async0() {
#if __has_builtin(__builtin_amdgcn_s_wait_asynccnt)
  __builtin_amdgcn_s_wait_asynccnt(0);
#elif defined(__AMDGCN__)
  asm volatile("s_wait_asynccnt 0" ::: "memory");
#endif
}

// Single-instruction hardware reciprocal / rsqrt (v_rcp_f32 / v_rsq_f32, ~1 ulp).
__device__ __forceinline__ float gs_rcp(float x) {
#if __has_builtin(__builtin_amdgcn_rcpf)
  return __builtin_amdgcn_rcpf(x);
#else
  return 1.0f / x;
#endif
}
__device__ __forceinline__ float gs_rsq(float x) {
#if __has_builtin(__builtin_amdgcn_rsqf)
  return __builtin_amdgcn_rsqf(x);
#else
  return rsqrtf(x);
#endif
}

#if GS_HAS_ASYNC_LDS
// IOFFSET is applied by HW to BOTH the global and the LDS address (ISA 08 §4.4),
// so one base pair + literal offsets covers the whole tile with no per-op ALU.
// Template recursion keeps the offset an integer constant expression.
template <int K> struct AsyncLoadTile {
  static __device__ __forceinline__ void run(glob_v4i* g, lds_v4i* l) {
    AsyncLoadTile<K - 1>::run(g, l);
    __builtin_amdgcn_global_load_async_to_lds_b128(g, l, (K - 1) * CHUNK_BYTES, 0);
  }
};
template <> struct AsyncLoadTile<0> {
  static __device__ __forceinline__ void run(glob_v4i*, lds_v4i*) {}
};

template <int K> struct AsyncStoreTile {
  static __device__ __forceinline__ void run(glob_v4i* g, lds_v4i* l) {
    AsyncStoreTile<K - 1>::run(g, l);
    __builtin_amdgcn_global_store_async_from_lds_b128(g, l, (K - 1) * CHUNK_BYTES, 0);
  }
};
template <> struct AsyncStoreTile<0> {
  static __device__ __forceinline__ void run(glob_v4i*, lds_v4i*) {}
};
#endif

// Per-matrix Gram-Schmidt, operating in-place on the LDS tile.
template <bool BOUNDED>
__device__ __forceinline__ void compute_tile(float* smem, int tid, int mat_base, int nmat) {
  #pragma unroll
  for (int k = 0; k < MATS_PER_THR; ++k) {
    const int lm = tid + k * THREADS;      // addr 36*lm B -> bank 9*lm mod 64: conflict-free
    if (!BOUNDED || (mat_base + lm) < nmat) {
      float* p = smem + lm * 9;
      const float ax = p[0], ay = p[1], az = p[2];
      const float bx = p[3], by = p[4], bz = p[5];
      const float cx = p[6], cy = p[7], cz = p[8];

      const float daa    = ax * ax + ay * ay + az * az;      // dot(ux,ux)
      const float inv_aa = gs_rcp(daa);                      // v_rcp_f32

      const float dab = ax * bx + ay * by + az * bz;         // dot(ux,vy)
      const float t1  = dab * inv_aa;
      const float ux = bx - t1 * ax, uy = by - t1 * ay, uz = bz - t1 * az;

      const float duu = ux * ux + uy * uy + uz * uz;         // dot(uy,uy)
      const float dac = ax * cx + ay * cy + az * cz;         // dot(ux,vz)
      const float duc = ux * cx + uy * cy + uz * cz;         // dot(uy,vz)
      const float t2  = dac * inv_aa;
      const float t3  = duc * gs_rcp(duu);                   // v_rcp_f32
      const float wx = cx - t2 * ax - t3 * ux;
      const float wy = cy - t2 * ay - t3 * uy;
      const float wz = cz - t2 * az - t3 * uz;

      const float dww = wx * wx + wy * wy + wz * wz;
      const float ra = gs_rsq(daa);                          // v_rsq_f32
      const float ru = gs_rsq(duu);
      const float rw = gs_rsq(dww);

      p[0] = ax * ra; p[1] = ay * ra; p[2] = az * ra;        // ex
      p[3] = ux * ru; p[4] = uy * ru; p[5] = uz * ru;        // ey
      p[6] = wx * rw; p[7] = wy * rw; p[8] = wz * rw;        // ez
    }
  }
}

__global__ __launch_bounds__(THREADS)
void gram_schmidt_3x3_kernel(const float* __restrict__ in,
                             float* __restrict__ out,
                             int nmat, long f4_total) {
  __shared__ float smem[TILE_FLOATS];  // 36 KB of the WGP's 320 KB

  const int  tid          = threadIdx.x;
  const long tile_base_f4 = (long)blockIdx.x * F4_PER_TILE;
  const int  mat_base     = blockIdx.x * TILE_MATS;
  const bool full_tile    = (tile_base_f4 + F4_PER_TILE) <= f4_total;  // uniform per block

  if (full_tile) {
    // ---------- fast path: straight-line, zero per-op guards ----------
    const long base_f = tile_base_f4 * 4 + (long)tid * 4;   // float index of this lane's base
#if GS_HAS_ASYNC_LDS
    AsyncLoadTile<F4_PER_THREAD>::run((glob_v4i*)(in + base_f),
                                      (lds_v4i*)(smem + tid * 4));
    gs_wait_async0();
#else
    #pragma unroll
    for (int k = 0; k < F4_PER_THREAD; ++k)
      *(float4*)(smem + (k * THREADS + tid) * 4) =
          *(const float4*)(in + base_f + (long)k * THREADS * 4);
#endif
    __syncthreads();

    compute_tile<false>(smem, tid, mat_base, nmat);
    __syncthreads();

#if GS_HAS_ASYNC_LDS
    AsyncStoreTile<F4_PER_THREAD>::run((glob_v4i*)(out + base_f),
                                       (lds_v4i*)(smem + tid * 4));
    gs_wait_async0();   // drain before S_ENDPGM / LDS reuse
#else
    #pragma unroll
    for (int k = 0; k < F4_PER_THREAD; ++k)
      *(float4*)(out + base_f + (long)k * THREADS * 4) =
          *(const float4*)(smem + (k * THREADS + tid) * 4);
#endif
  } else {
    // ---------- tail path: per-element guards (generic) ----------
    #pragma unroll
    for (int k = 0; k < F4_PER_THREAD; ++k) {
      const int  lf4 = k * THREADS + tid;
      const long g   = tile_base_f4 + lf4;
      if (g < f4_total) {
#if GS_HAS_ASYNC_LDS
        __builtin_amdgcn_global_load_async_to_lds_b128(
            (glob_v4i*)((const float4*)in + g), (lds_v4i*)(smem + lf4 * 4), 0, 0);
#else
        *(float4*)(smem + lf4 * 4) = ((const float4*)in)[g];
#endif
      }
    }
#if GS_HAS_ASYNC_LDS
    gs_wait_async0();
#endif
    __syncthreads();

    compute_tile<true>(smem, tid, mat_base, nmat);
    __syncthreads();

    #pragma unroll
    for (int k = 0; k < F4_PER_THREAD; ++k) {
      const int  lf4 = k * THREADS + tid;
      const long g   = tile_base_f4 + lf4;
      if (g < f4_total) {
#if GS_HAS_ASYNC_LDS
        __builtin_amdgcn_global_store_async_from_lds_b128(
            (glob_v4i*)((float4*)out + g), (lds_v4i*)(smem + lf4 * 4), 0, 0);
#else
        ((float4*)out)[g] = *(const float4*)(smem + lf4 * 4);
#endif
      }
    }
#if GS_HAS_ASYNC_LDS
    gs_wait_async0();
#endif
  }
}

extern "C" void kernel_launch(void* const* d_in, const int* in_sizes, int n_in,
                              void* d_out, int out_size, void* d_ws, size_t ws_size,
                              hipStream_t stream) {
  (void)n_in; (void)out_size; (void)d_ws; (void)ws_size;
  const float* in  = (const float*)d_in[0];
  float*       out = (float*)d_out;

  const long n_floats = (long)in_sizes[0];   // 4194304 * 3 * 3
  const int  nmat     = (int)(n_floats / 9);
  const long f4_total = n_floats / 4;

  const int blocks = (nmat + TILE_MATS - 1) / TILE_MATS;  // 4096, all full tiles
  gram_schmidt_3x3_kernel<<<blocks, THREADS, 0, stream>>>(in, out, nmat, f4_total);
}